// MaxUnpooling2D_4123168604810
// MI455X (gfx1250) — compile-verified
//
#include <hip/hip_runtime.h>
#include <hip/hip_bf16.h>

// MaxUnpooling2D, B=16 H=64 W=64 C=256, 2x2 windows -> out 16x128x128x256.
// Gather-style formulation: each thread owns one input vec4 (b,h,w,c..c+3)
// and writes the 4 output cells of its 2x2 window (update where mask matches,
// else 0). Every output cell is written exactly once -> no zero pass, no
// atomics. Traffic = 64 MiB (updates) + 64 MiB (mask) + 256 MiB (out) =
// 384 MiB, the roofline floor (~16.5 us at 23.3 TB/s HBM).
//
// CDNA5 specifics: wave32 lanes cover consecutive channel vec4s so every
// global_load_b128 / global_store_b128 is a contiguous 512B burst per wave;
// nontemporal builtins emit the gfx1250 TH=NT cache-control bits since the
// 256 MiB output exceeds the 192 MB L2 and nothing is re-read.

typedef float v4f __attribute__((ext_vector_type(4)));
typedef int   v4i __attribute__((ext_vector_type(4)));

namespace {
constexpr int kB  = 16;
constexpr int kH  = 64;
constexpr int kW  = 64;
constexpr int kC  = 256;
constexpr int kHo = kH * 2;
constexpr int kWo = kW * 2;
constexpr int kRowStride = kWo * kC;               // 32768 floats per output row
constexpr size_t kBatchStride = (size_t)kHo * kWo * kC;  // 4,194,304 floats
}

__global__ __launch_bounds__(256) void MaxUnpooling2D_kernel(
    const float* __restrict__ updates,
    const int*   __restrict__ mask,
    float*       __restrict__ out,
    int n_vec4)
{
    const int i4 = blockIdx.x * blockDim.x + threadIdx.x;  // input vec4 id
    if (i4 >= n_vec4) return;

    // Decode (b, h, w, c) — all power-of-two dims, pure shifts.
    const int c = (i4 & 63) << 2;        // channel, step 4  (C/4 = 64)
    const int w = (i4 >> 6) & 63;
    const int h = (i4 >> 12) & 63;
    const int b = i4 >> 18;

    const size_t in_off = (size_t)i4 * 4;
    const v4f u = __builtin_nontemporal_load((const v4f*)(updates + in_off));
    const v4i m = __builtin_nontemporal_load((const v4i*)(mask + in_off));

    // Within-batch flat index of output cell (yo=2h, xo=2w, c):
    //   t00 = (2h*Wo + 2w)*C + c = h*65536 + w*512 + c
    const int t00 = h * (2 * kRowStride) + w * (2 * kC) + c;
    const size_t base = (size_t)b * kBatchStride + (size_t)t00;

    // Four window positions: (r,s) offsets {0,Wo*C} x {0,C}.
    #pragma unroll
    for (int r = 0; r < 2; ++r) {
        #pragma unroll
        for (int s = 0; s < 2; ++s) {
            const int t = t00 + r * kRowStride + s * kC;
            v4f v;
            v.x = (m.x == t + 0) ? u.x : 0.0f;
            v.y = (m.y == t + 1) ? u.y : 0.0f;
            v.z = (m.z == t + 2) ? u.z : 0.0f;
            v.w = (m.w == t + 3) ? u.w : 0.0f;
            __builtin_nontemporal_store(
                v, (v4f*)(out + base + (size_t)r * kRowStride + (size_t)s * kC));
        }
    }
}

extern "C" void kernel_launch(void* const* d_in, const int* in_sizes, int n_in,
                              void* d_out, int out_size, void* d_ws, size_t ws_size,
                              hipStream_t stream) {
    (void)n_in; (void)out_size; (void)d_ws; (void)ws_size;
    const float* updates = (const float*)d_in[0];
    const int*   mask    = (const int*)d_in[1];
    float*       out     = (float*)d_out;

    const int n = in_sizes[0];          // 16*64*64*256 = 16,777,216 elements
    const int n_vec4 = n / 4;           // 4,194,304 threads
    const int block = 256;              // 8 waves (wave32) per workgroup
    const int grid = (n_vec4 + block - 1) / block;

    MaxUnpooling2D_kernel<<<grid, block, 0, stream>>>(updates, mask, out, n_vec4);
}